// mHCConnection_61100204753729
// MI455X (gfx1250) — compile-verified
//
#include <hip/hip_runtime.h>
#include <math.h>

#define ND 4096
#define DD 1024
#define NN 4
#define ROWS 16
#define WAVES 8
#define KSLICE (ND / WAVES)   /* 512 */
#define CLAMP_V 1e4f
#define EPS_SK 1e-8f
#define EPS_RMS 1e-6f

typedef __attribute__((ext_vector_type(2))) float v2f;
typedef __attribute__((ext_vector_type(8))) float v8f;

// clip(nan_to_num(x), -1e4, 1e4):  +-inf already clamp to +-1e4 via min/max,
// so only the NaN case needs special handling -> ~3 VALU per element.
__device__ __forceinline__ float clipn(float v) {
    float c = fminf(fmaxf(v, -CLAMP_V), CLAMP_V);   // v_med3_num_f32
    return (v != v) ? 0.f : c;
}
// exact nan_to_num (finite values beyond +-1e4 pass through)
__device__ __forceinline__ float nan2num(float v) {
    v = (v != v) ? 0.f : v;
    if (v ==  __builtin_inff()) v =  CLAMP_V;
    if (v == -__builtin_inff()) v = -CLAMP_V;
    return v;
}
__device__ __forceinline__ float sigmoidf(float z) {
    return 1.f / (1.f + __expf(-z));
}

// ---------------- prep: Wc[o][k] = W[o][k] * norm_weight[k], o in 0..31 ----------------
// rows 0..15  = W_res   (h_res outputs, o = n*4+m)
// rows 16..19 = W_post  (h_post outputs)
// rows 20..31 = 0       (padding for second WMMA N-tile)
__global__ void prep_wc(const float* __restrict__ Wpost,
                        const float* __restrict__ Wres,
                        const float* __restrict__ wnorm,
                        float* __restrict__ Wc) {
    int idx = blockIdx.x * blockDim.x + threadIdx.x;
    if (idx >= 32 * ND) return;
    int o = idx >> 12;
    int k = idx & (ND - 1);
    float v = 0.f;
    if (o < 16)       v = Wres[o * ND + k] * wnorm[k];
    else if (o < 20)  v = Wpost[(o - 16) * ND + k] * wnorm[k];
    Wc[idx] = v;
}

// ---------------- main fused kernel: one workgroup (8 wave32) per 16 rows ----------------
__global__ __launch_bounds__(256) void mhc_main(
    const float* __restrict__ x,        // [BT][4096]
    const float* __restrict__ L,        // [BT][1024]
    const float* __restrict__ Wc,       // [32][4096]
    const float* __restrict__ bias_post,// [4]
    const float* __restrict__ bias_res, // [16]
    const float* __restrict__ p_apost,
    const float* __restrict__ p_ares,
    const float* __restrict__ p_slogit,
    float* __restrict__ out,            // [BT][4096]
    int BT)
{
    __shared__ float s_hred[2][WAVES][256]; // per-wave WMMA partials (2 N-tiles)
    __shared__ float s_h[2][256];           // reduced raw dots
    __shared__ float s_ssred[WAVES][ROWS];  // per-wave sum-of-squares partials
    __shared__ float s_C[ROWS][NN][NN];     // alpha * H_res
    __shared__ float s_wc[ROWS][NN];        // (1-alpha) * H_post_norm

    const int tid  = threadIdx.x;
    const int w    = tid >> 5;
    const int lane = tid & 31;
    const int blockRow = blockIdx.x * ROWS;

    // ---------- Phase 1: skinny GEMM via V_WMMA_F32_16X16X4_F32 + fused sumsq ----------
    // A (16x4 f32): lanes 0-15 = rows M with K{0,1}; lanes 16-31 = rows with K{2,3}
    // B (4x16 f32): lanes 0-15 = cols N with K{0,1}; lanes 16-31 = cols with K{2,3}
    // Each (row,k) element of the 16x4096 A tile is touched exactly once across the
    // workgroup, so per-lane a.x^2+a.y^2 accumulation reduces to exact sumsq[row].
    v8f acc0 = {0.f,0.f,0.f,0.f,0.f,0.f,0.f,0.f};
    v8f acc1 = {0.f,0.f,0.f,0.f,0.f,0.f,0.f,0.f};
    float ss = 0.f;
    {
        int arow = blockRow + (lane & 15);
        if (arow >= BT) arow = BT - 1;            // clamp (pad rows' results unused)
        const int koff = (lane >> 4) * 2;
        const float* xa  = x  + (size_t)arow * ND + koff;
        const float* b0p = Wc + (size_t)(lane & 15) * ND + koff;
        const float* b1p = Wc + (size_t)(16 + (lane & 15)) * ND + koff;
        const int kbeg = w * KSLICE, kend = kbeg + KSLICE;
        for (int k0 = kbeg; k0 < kend; k0 += 4) {
            v2f av = *(const v2f*)(xa + k0);
            v2f a;
            a.x = clipn(av.x);
            a.y = clipn(av.y);
            ss = fmaf(a.x, a.x, ss);
            ss = fmaf(a.y, a.y, ss);
            v2f b0 = *(const v2f*)(b0p + k0);
            v2f b1 = *(const v2f*)(b1p + k0);
            acc0 = __builtin_amdgcn_wmma_f32_16x16x4_f32(false, a, false, b0,
                                                         (short)0, acc0, false, false);
            acc1 = __builtin_amdgcn_wmma_f32_16x16x4_f32(false, a, false, b1,
                                                         (short)0, acc1, false, false);
        }
    }
    // fold the two K-halves (lane, lane+16) together; lanes 0-15 then hold row sums
    ss += __shfl_xor(ss, 16, 32);
    if (lane < ROWS) s_ssred[w][lane] = ss;

    // D layout: VGPR j, lanes 0-15 -> M=j ; lanes 16-31 -> M=j+8 ; N = lane&15
    #pragma unroll
    for (int j = 0; j < 8; ++j) {
        int M  = (lane < 16) ? j : (j + 8);
        int Nc = lane & 15;
        s_hred[0][w][M * 16 + Nc] = acc0[j];
        s_hred[1][w][M * 16 + Nc] = acc1[j];
    }
    __syncthreads();

    // cross-wave reduction of the two 16x16 tiles
    {
        float t0 = 0.f, t1 = 0.f;
        #pragma unroll
        for (int ww = 0; ww < WAVES; ++ww) {
            t0 += s_hred[0][ww][tid];
            t1 += s_hred[1][ww][tid];
        }
        s_h[0][tid] = t0;   // raw h_res dots   (row*16 + (n*4+m))
        s_h[1][tid] = t1;   // raw h_post dots  (row*16 + n, n<4 valid)
    }
    __syncthreads();

    // ---------- Phase 2: per-row scalar math (norm, H_post_norm, Sinkhorn) ----------
    if (tid < ROWS) {
        const int r = tid;
        float sumsq = 0.f;
        #pragma unroll
        for (int ww = 0; ww < WAVES; ++ww) sumsq += s_ssred[ww][r];

        const float a_post = fminf(fmaxf(p_apost[0], 0.f), 0.1f);
        const float a_res  = fminf(fmaxf(p_ares[0],  0.f), 0.1f);
        const float alpha  = sigmoidf(p_slogit[0]);
        const float inv_norm = 1.f / sqrtf(sumsq * (1.f / (float)ND) + EPS_RMS);

        // H_post_norm
        float Hp[NN]; float hsum = 0.f;
        #pragma unroll
        for (int n = 0; n < NN; ++n) {
            float h = s_h[1][r * 16 + n] * inv_norm;
            float z = fminf(fmaxf(fmaf(a_post, h, bias_post[n]), -10.f), 10.f);
            Hp[n] = 2.f * sigmoidf(z);
            hsum += Hp[n];
        }
        const float hinv = 1.f / (hsum + EPS_SK);
        #pragma unroll
        for (int n = 0; n < NN; ++n) s_wc[r][n] = (1.f - alpha) * Hp[n] * hinv;

        // Sinkhorn-Knopp on 4x4, 20 iterations
        float M4[NN][NN];
        #pragma unroll
        for (int n = 0; n < NN; ++n)
            #pragma unroll
            for (int m = 0; m < NN; ++m)
                M4[n][m] = __expf(fmaf(a_res, s_h[0][r * 16 + n * 4 + m] * inv_norm,
                                       bias_res[n * 4 + m]));
        for (int it = 0; it < 20; ++it) {
            #pragma unroll
            for (int n = 0; n < NN; ++n) {
                float rs = M4[n][0] + M4[n][1] + M4[n][2] + M4[n][3] + EPS_SK;
                float ri = 1.f / rs;
                #pragma unroll
                for (int m = 0; m < NN; ++m) M4[n][m] *= ri;
            }
            #pragma unroll
            for (int m = 0; m < NN; ++m) {
                float cs = M4[0][m] + M4[1][m] + M4[2][m] + M4[3][m] + EPS_SK;
                float ci = 1.f / cs;
                #pragma unroll
                for (int n = 0; n < NN; ++n) M4[n][m] *= ci;
            }
        }
        #pragma unroll
        for (int n = 0; n < NN; ++n)
            #pragma unroll
            for (int m = 0; m < NN; ++m)
                s_C[r][n][m] = alpha * M4[n][m];
    }
    __syncthreads();

    // ---------- Phase 3: streamed output  out = C @ x3 + wc * L ----------
    #pragma unroll
    for (int rr = 0; rr < 2; ++rr) {
        int r = 2 * w + rr;
        int R = blockRow + r;
        if (R >= BT) continue;
        const float* xr   = x   + (size_t)R * ND;
        const float* Lr   = L   + (size_t)R * DD;
        float*       outr = out + (size_t)R * ND;
        float c[NN][NN], wcf[NN];
        #pragma unroll
        for (int n = 0; n < NN; ++n) {
            wcf[n] = s_wc[r][n];
            #pragma unroll
            for (int m = 0; m < NN; ++m) c[n][m] = s_C[r][n][m];
        }
        for (int it = 0; it < DD / 128; ++it) {   // 8 iters of float4 per lane
            int d = (it * 32 + lane) * 4;
            float4 l4 = *(const float4*)(Lr + d);
            float lv[4] = {l4.x, l4.y, l4.z, l4.w};
            float xm[NN][4];
            #pragma unroll
            for (int m = 0; m < NN; ++m) {
                float4 v = *(const float4*)(xr + m * DD + d);
                xm[m][0] = nan2num(v.x); xm[m][1] = nan2num(v.y);
                xm[m][2] = nan2num(v.z); xm[m][3] = nan2num(v.w);
            }
            #pragma unroll
            for (int n = 0; n < NN; ++n) {
                float o[4];
                #pragma unroll
                for (int q = 0; q < 4; ++q) {
                    float acc = wcf[n] * lv[q];
                    #pragma unroll
                    for (int m = 0; m < NN; ++m) acc = fmaf(c[n][m], xm[m][q], acc);
                    o[q] = acc;
                }
                float4 o4 = {o[0], o[1], o[2], o[3]};
                *(float4*)(outr + n * DD + d) = o4;
            }
        }
    }
}

extern "C" void kernel_launch(void* const* d_in, const int* in_sizes, int n_in,
                              void* d_out, int out_size, void* d_ws, size_t ws_size,
                              hipStream_t stream) {
    const float* x         = (const float*)d_in[0];
    const float* L         = (const float*)d_in[1];
    // d_in[2] = W_pre (unused: x_single is not part of the output)
    const float* Wpost     = (const float*)d_in[3];
    const float* Wres      = (const float*)d_in[4];
    // d_in[5] = bias_pre (unused)
    const float* bias_post = (const float*)d_in[6];
    const float* bias_res  = (const float*)d_in[7];
    // d_in[8] = alpha_pre (unused)
    const float* apost     = (const float*)d_in[9];
    const float* ares      = (const float*)d_in[10];
    const float* slog      = (const float*)d_in[11];
    const float* wnorm     = (const float*)d_in[12];

    float* Wc = (float*)d_ws;                 // 32*4096 floats = 512 KB scratch
    const int BT = in_sizes[0] / ND;          // 8192

    prep_wc<<<(32 * ND + 255) / 256, 256, 0, stream>>>(Wpost, Wres, wnorm, Wc);

    const int blocks = (BT + ROWS - 1) / ROWS; // 512
    mhc_main<<<blocks, 256, 0, stream>>>(x, L, Wc, bias_post, bias_res,
                                         apost, ares, slog, (float*)d_out, BT);
}